// SparseLinear_57380763075145
// MI455X (gfx1250) — compile-verified
//
#include <hip/hip_runtime.h>
#include <stdint.h>

typedef unsigned int u32;
typedef float v2f __attribute__((ext_vector_type(2)));
typedef float v8f __attribute__((ext_vector_type(8)));
typedef int   async_v4i __attribute__((vector_size(16)));   // matches builtin param

#define D_IN   4096
#define D_OUT  4096
#define TOKENS 4096
#define NW     (D_IN * D_OUT)        // 16,777,216 weights
#define K_RANK 15099493u             // floor(0.9 * (NW-1)), 0-indexed ascending order statistic

#if defined(__has_builtin)
#  if __has_builtin(__builtin_amdgcn_global_load_async_to_lds_b128)
#    define HAVE_ASYNC_LDS 1
#  endif
#endif

#ifdef HAVE_ASYNC_LDS
__device__ __forceinline__ void async_copy_b128(const float* __restrict__ g, float* l) {
    __builtin_amdgcn_global_load_async_to_lds_b128(
        (__attribute__((address_space(1))) async_v4i*)(unsigned long long)(const void*)g,
        (__attribute__((address_space(3))) async_v4i*)l,
        /*offset=*/0, /*cpol=*/0);
}
#endif

__device__ __forceinline__ void wait_async0() {
#if defined(__has_builtin)
#  if __has_builtin(__builtin_amdgcn_s_wait_asynccnt)
    __builtin_amdgcn_s_wait_asynccnt(0);
#  else
    asm volatile("s_wait_asynccnt 0x0" ::: "memory");
#  endif
#else
    asm volatile("s_wait_asynccnt 0x0" ::: "memory");
#endif
}

// ---------------------------------------------------------------------------
// Phase 1: exact k-th order statistic of |W| (bit pattern) via 3-level radix
// count: bits [30:19] (4096 bins), [18:7] (4096 bins), [6:0] (128 bins).
// ---------------------------------------------------------------------------

__global__ void zero_bins(u32* __restrict__ bins) {
    int i = blockIdx.x * blockDim.x + threadIdx.x;
    if (i < 4096) bins[i] = 0u;
}

__global__ __launch_bounds__(256)
void hist_kernel(const u32* __restrict__ wbits, u32* __restrict__ bins,
                 const u32* __restrict__ state, int level) {
    __shared__ u32 h[4096];
    const int nb = (level == 2) ? 128 : 4096;
    for (int i = threadIdx.x; i < 4096; i += blockDim.x) h[i] = 0u;
    __syncthreads();

    const u32 pref = (level > 0) ? state[0] : 0u;  // uniform -> scalar load
    const uint4* w4 = (const uint4*)wbits;
    const u32 tid = blockIdx.x * blockDim.x + threadIdx.x;
    const u32 stride = gridDim.x * blockDim.x;

    for (u32 i4 = tid; i4 < (NW / 4); i4 += stride) {
        uint4 v = w4[i4];
        u32 e[4] = {v.x, v.y, v.z, v.w};
        #pragma unroll
        for (int c = 0; c < 4; ++c) {
            u32 u = e[c] & 0x7fffffffu;
            bool ok; u32 b;
            if (level == 0)      { ok = true;                  b = u >> 19;          }
            else if (level == 1) { ok = ((u >> 19) == pref);   b = (u >> 7) & 0xFFFu;}
            else                 { ok = ((u >> 7)  == pref);   b = u & 0x7Fu;        }
            if (ok) atomicAdd(&h[b], 1u);
        }
    }
    __syncthreads();
    for (int i = threadIdx.x; i < nb; i += blockDim.x)
        if (h[i]) atomicAdd(&bins[i], h[i]);
}

__global__ __launch_bounds__(256)
void select_kernel(const u32* __restrict__ bins, u32* __restrict__ state, int level) {
    __shared__ u32 csum[256];
    const int nb = (level == 2) ? 128 : 4096;
    const int chunk = (nb + 255) / 256;              // 16 or 1
    u32 s = 0;
    const int base = threadIdx.x * chunk;
    for (int i = 0; i < chunk; ++i) {
        int idx = base + i;
        if (idx < nb) s += bins[idx];
    }
    csum[threadIdx.x] = s;
    __syncthreads();
    if (threadIdx.x == 0) {
        u32 rank   = (level == 0) ? K_RANK : state[1];
        u32 prefix = (level == 0) ? 0u     : state[0];
        u32 cum = 0;
        int c = 0;
        while (c + 1 < 256 && cum + csum[c] <= rank) { cum += csum[c]; ++c; }
        int b = c * chunk;
        while (b + 1 < nb && cum + bins[b] <= rank) { cum += bins[b]; ++b; }
        const int bits = (level == 2) ? 7 : 12;
        state[0] = (prefix << bits) | (u32)b;        // after level 2: full 31-bit |w| pattern
        state[1] = rank - cum;
    }
}

// ---------------------------------------------------------------------------
// Phase 2: y = x @ prune(W)^T + bias via V_WMMA_F32_16X16X4_F32.
// 256 threads (8 waves), 128x128 tile, wave tile 32x64, K staged 32-wide in
// double-buffered LDS. X tiles stream via global_load_async_to_lds_b128
// (ASYNCcnt); W tiles go through VGPRs (prune mask) issued before compute so
// their latency hides under the 64-WMMA stream.
// ---------------------------------------------------------------------------

#define TILE_M 128
#define TILE_N 128
#define TILE_K 32
#define LDSP   36

__global__ __launch_bounds__(256)
void spgemm_kernel(const float* __restrict__ X, const float* __restrict__ W,
                   const float* __restrict__ bias, const u32* __restrict__ state,
                   float* __restrict__ Y) {
    __shared__ __align__(16) float xs[2][TILE_M * LDSP];
    __shared__ __align__(16) float wsm[2][TILE_N * LDSP];

    const float thr = __uint_as_float(state[0]);

    const int tid  = threadIdx.x;
    const int lane = tid & 31;
    const int l    = lane & 15;     // N / M index within 16
    const int hh   = lane >> 4;     // lane-half: selects K pair / M upper half
    const int wv   = tid >> 5;      // wave 0..7
    const int wrow = wv >> 1;       // 0..3 -> 32-row band of M
    const int wcol = wv & 1;        // 0..1 -> 64-col band of N

    const int m0 = blockIdx.y * TILE_M;
    const int n0 = blockIdx.x * TILE_N;

    // tile loader mapping: 8 threads/row * float4, 32 rows per sweep, 4 sweeps
    const int lr = tid >> 3;        // 0..31
    const int lc = (tid & 7) * 4;   // 0,4,...,28

    v8f acc[2][4];
    #pragma unroll
    for (int i = 0; i < 2; ++i)
        #pragma unroll
        for (int j = 0; j < 4; ++j)
            #pragma unroll
            for (int q = 0; q < 8; ++q) acc[i][j][q] = 0.0f;

    auto load_x_tile = [&](int buf, int kt) {
        #pragma unroll
        for (int p = 0; p < 4; ++p) {
            const int r = lr + p * 32;
#ifdef HAVE_ASYNC_LDS
            async_copy_b128(&X[(size_t)(m0 + r) * D_IN + kt + lc],
                            &xs[buf][r * LDSP + lc]);
#else
            *(float4*)&xs[buf][r * LDSP + lc] =
                *(const float4*)&X[(size_t)(m0 + r) * D_IN + kt + lc];
#endif
        }
    };
    auto load_w_regs = [&](int kt, float4* wreg) {
        #pragma unroll
        for (int p = 0; p < 4; ++p)
            wreg[p] = *(const float4*)&W[(size_t)(n0 + lr + p * 32) * D_IN + kt + lc];
    };
    auto store_w_tile = [&](int buf, const float4* wreg) {
        #pragma unroll
        for (int p = 0; p < 4; ++p) {
            float4 v = wreg[p];
            v.x = (__builtin_fabsf(v.x) > thr) ? v.x : 0.0f;
            v.y = (__builtin_fabsf(v.y) > thr) ? v.y : 0.0f;
            v.z = (__builtin_fabsf(v.z) > thr) ? v.z : 0.0f;
            v.w = (__builtin_fabsf(v.w) > thr) ? v.w : 0.0f;
            *(float4*)&wsm[buf][(lr + p * 32) * LDSP + lc] = v;
        }
    };

    // ---- prologue: stage tile 0 ----
    float4 wreg[4];
    load_x_tile(0, 0);
    load_w_regs(0, wreg);
    store_w_tile(0, wreg);
#ifdef HAVE_ASYNC_LDS
    wait_async0();
#endif
    __syncthreads();

    int cur = 0;
    for (int kt = 0; kt < D_IN; kt += TILE_K) {
        const int nxt = cur ^ 1;
        const bool has_next = (kt + TILE_K) < D_IN;
        if (has_next) {
            load_x_tile(nxt, kt + TILE_K);          // async -> LDS, overlaps compute
            load_w_regs(kt + TILE_K, wreg);         // global loads in flight during WMMAs
            if (kt + 2 * TILE_K < D_IN) {           // warm L2 two tiles ahead
                __builtin_prefetch(&X[(size_t)(m0 + lr) * D_IN + kt + 2 * TILE_K + lc], 0, 1);
                __builtin_prefetch(&W[(size_t)(n0 + lr) * D_IN + kt + 2 * TILE_K + lc], 0, 1);
            }
        }

        const float* xb = xs[cur];
        const float* wb = wsm[cur];
        #pragma unroll
        for (int kk = 0; kk < TILE_K; kk += 4) {
            const int kc = kk + 2 * hh;   // this lane-half's K pair
            v2f a0 = *(const v2f*)&xb[(wrow * 32      + l) * LDSP + kc];
            v2f a1 = *(const v2f*)&xb[(wrow * 32 + 16 + l) * LDSP + kc];
            v2f b0 = *(const v2f*)&wb[(wcol * 64      + l) * LDSP + kc];
            v2f b1 = *(const v2f*)&wb[(wcol * 64 + 16 + l) * LDSP + kc];
            v2f b2 = *(const v2f*)&wb[(wcol * 64 + 32 + l) * LDSP + kc];
            v2f b3 = *(const v2f*)&wb[(wcol * 64 + 48 + l) * LDSP + kc];

            acc[0][0] = __builtin_amdgcn_wmma_f32_16x16x4_f32(false, a0, false, b0, (short)0, acc[0][0], false, false);
            acc[0][1] = __builtin_amdgcn_wmma_f32_16x16x4_f32(false, a0, false, b1, (short)0, acc[0][1], false, false);
            acc[0][2] = __builtin_amdgcn_wmma_f32_16x16x4_f32(false, a0, false, b2, (short)0, acc[0][2], false, false);
            acc[0][3] = __builtin_amdgcn_wmma_f32_16x16x4_f32(false, a0, false, b3, (short)0, acc[0][3], false, false);
            acc[1][0] = __builtin_amdgcn_wmma_f32_16x16x4_f32(false, a1, false, b0, (short)0, acc[1][0], false, false);
            acc[1][1] = __builtin_amdgcn_wmma_f32_16x16x4_f32(false, a1, false, b1, (short)0, acc[1][1], false, false);
            acc[1][2] = __builtin_amdgcn_wmma_f32_16x16x4_f32(false, a1, false, b2, (short)0, acc[1][2], false, false);
            acc[1][3] = __builtin_amdgcn_wmma_f32_16x16x4_f32(false, a1, false, b3, (short)0, acc[1][3], false, false);
        }

        if (has_next) store_w_tile(nxt, wreg);      // mask + ds_store for next tile
#ifdef HAVE_ASYNC_LDS
        wait_async0();                              // async X copy landed
#endif
        __syncthreads();
        cur = nxt;
    }

    // Epilogue: C/D layout — VGPR v holds M = v + 8*hh, N = l within each 16x16 tile.
    #pragma unroll
    for (int ti = 0; ti < 2; ++ti) {
        #pragma unroll
        for (int tj = 0; tj < 4; ++tj) {
            const int col = n0 + wcol * 64 + tj * 16 + l;
            const float bv = bias[col];
            #pragma unroll
            for (int v = 0; v < 8; ++v) {
                const int row = m0 + wrow * 32 + ti * 16 + hh * 8 + v;
                Y[(size_t)row * D_OUT + col] = acc[ti][tj][v] + bv;
            }
        }
    }
}

// ---------------------------------------------------------------------------

extern "C" void kernel_launch(void* const* d_in, const int* in_sizes, int n_in,
                              void* d_out, int out_size, void* d_ws, size_t ws_size,
                              hipStream_t stream) {
    const float* x    = (const float*)d_in[0];
    const float* w    = (const float*)d_in[1];
    const float* bias = (const float*)d_in[2];
    const u32* wbits  = (const u32*)d_in[1];
    float* y = (float*)d_out;

    u32* bins  = (u32*)d_ws;       // 4096 bins
    u32* state = bins + 4096;      // [0]=prefix/threshold bits, [1]=residual rank

    for (int level = 0; level < 3; ++level) {
        zero_bins<<<16, 256, 0, stream>>>(bins);
        hist_kernel<<<2048, 256, 0, stream>>>(wbits, bins, state, level);
        select_kernel<<<1, 256, 0, stream>>>(bins, state, level);
    }

    dim3 grid(D_OUT / TILE_N, TOKENS / TILE_M);
    spgemm_kernel<<<grid, 256, 0, stream>>>(x, w, bias, state, y);
}